// LASMConvssw_29197187678930
// MI455X (gfx1250) — compile-verified
//
#include <hip/hip_runtime.h>
#include <math.h>

// Problem constants from the reference
#define BB  4
#define PP  20000
#define NBR 9
#define WN  17
#define CI  64
#define CO  128

#define PTS 32     // points per block (block covers 128 rows = 4 batches x 32 points)
#define KCH 16     // CI chunk staged per gather pass
#define AST 20     // LDS row stride (floats) for fuse tile  (16 rows -> 16 distinct banks, 8B aligned)
#define BST 144    // LDS row stride (floats) for weight tile (half-waves hit disjoint bank sets)

typedef __attribute__((ext_vector_type(2))) float v2f;
typedef __attribute__((ext_vector_type(8))) float v8f;

#if defined(__has_builtin)
#if __has_builtin(__builtin_amdgcn_global_load_async_to_lds_b128) && \
    __has_builtin(__builtin_amdgcn_s_wait_asynccnt)
#define USE_ASYNC_LDS 1
#endif
#endif
#ifndef USE_ASYNC_LDS
#define USE_ASYNC_LDS 0
#endif

#if USE_ASYNC_LDS
typedef int a_v4i __attribute__((vector_size(4 * sizeof(int))));
typedef __attribute__((address_space(1))) a_v4i* gp_v4i;   // global int4*
typedef __attribute__((address_space(3))) a_v4i* lp_v4i;   // LDS int4*
#endif

__launch_bounds__(256)
__global__ void lasm_conv_wmma_kernel(const float* __restrict__ in_pc,
                                      const float* __restrict__ raw_w,
                                      const float* __restrict__ weights,
                                      const float* __restrict__ bias,
                                      const int*   __restrict__ nbr,
                                      float*       __restrict__ out) {
  __shared__ __align__(16) int   s_nid[PTS * NBR];            // clamped neighbor ids      (1.1 KB)
  __shared__ __align__(16) float s_w[PTS * NBR * WN];         // masked w_weights          (19.1 KB)
  __shared__ __align__(16) float s_xg[BB * PTS * NBR * KCH];  // gathered neighbor feats   (72 KB)
  __shared__ __align__(16) float s_a[4 * PTS * AST];          // fuse tile: 128 rows x KCH (10 KB)
  __shared__ __align__(16) float s_b[KCH * BST];              // weight tile: KCH x CO     (9 KB)

  const int tid  = threadIdx.x;
  const int lane = tid & 31;
  const int wave = tid >> 5;
  const int rsg  = wave >> 1;        // row supergroup == batch index (0..3), 32 rows each
  const int ch   = wave & 1;         // column half (0..1), 64 cols each
  const int pt0  = blockIdx.x * PTS;

  // ---- stage clamped neighbor ids ----
  for (int e = tid; e < PTS * NBR; e += 256) {
    int id = nbr[pt0 * NBR + e];
    s_nid[e] = (id < PP) ? id : 0;
  }
  // ---- stage masked w_weights: s_w[pl][n][m] ----
  for (int e = tid; e < PTS * NBR * WN; e += 256) {
    int m  = e % WN;
    int n  = (e / WN) % NBR;
    int pl = e / (WN * NBR);
    int id = nbr[(pt0 + pl) * NBR + n];
    float w = raw_w[(pt0 + pl) * (NBR * WN) + n * WN + m];
    s_w[e] = (id < PP) ? w : 0.0f;
  }
  __syncthreads();

  v8f acc[2][4] = {};                // 32 rows x 64 cols per wave (2 M-tiles x 4 N-tiles)

  const int lm = lane & 15;
  const int kh = (lane >> 4) * 2;    // A/B fragment K-half select: 0 or 2

  for (int ih = 0; ih < CI / KCH; ++ih) {
    // ---- gather stage: s_xg[b][pl][n][0..KCH) = in_pc[b][nid][ih*KCH ..] ----
    for (int e = tid; e < BB * PTS * NBR * (KCH / 4); e += 256) {
      int i4 = e & 3;
      int n  = (e >> 2) % NBR;
      int pl = ((e >> 2) / NBR) & (PTS - 1);
      int b  = e / (4 * NBR * PTS);
      int id = s_nid[pl * NBR + n];
      const float* gp = in_pc + ((b * PP + id) * CI) + ih * KCH + i4 * 4;
      float*       lp = &s_xg[(((b * PTS + pl) * NBR + n) * KCH) + i4 * 4];
#if USE_ASYNC_LDS
      __builtin_amdgcn_global_load_async_to_lds_b128(
          (gp_v4i)(unsigned long long)gp, (lp_v4i)lp, 0, 0);
#else
      *(float4*)lp = *(const float4*)gp;
#endif
    }
#if USE_ASYNC_LDS
    __builtin_amdgcn_s_wait_asynccnt(0);
#endif

    for (int m = 0; m < WN; ++m) {
      __syncthreads();   // protect s_b/s_a readers of previous iteration; fence gather writes

      // ---- stage weight tile: s_b[k][o] = Wm[m][o][ih*KCH + k] (coalesced 16B/lane) ----
      for (int e = tid; e < (KCH / 4) * CO; e += 256) {
        int k4 = (e & 3) * 4;
        int o  = e >> 2;
        const float4 v = *(const float4*)(weights + m * (CO * CI) + o * CI + ih * KCH + k4);
        s_b[(k4 + 0) * BST + o] = v.x;
        s_b[(k4 + 1) * BST + o] = v.y;
        s_b[(k4 + 2) * BST + o] = v.z;
        s_b[(k4 + 3) * BST + o] = v.w;
      }

      // ---- fuse tile: s_a[r][k] = sum_n s_w[pl][n][m] * s_xg[b][pl][n][k] ----
      {
        int r  = tid >> 1;            // 0..127
        int kq = (tid & 1) * 8;       // 0 or 8 (two float4s each)
        int pl = r & (PTS - 1);
        int b  = r >> 5;
        float a0 = 0.f, a1 = 0.f, a2 = 0.f, a3 = 0.f;
        float a4 = 0.f, a5 = 0.f, a6 = 0.f, a7 = 0.f;
        #pragma unroll
        for (int n = 0; n < NBR; ++n) {
          float wv = s_w[(pl * NBR + n) * WN + m];
          const float* xp = &s_xg[(((b * PTS + pl) * NBR + n) * KCH) + kq];
          const float4 x0 = *(const float4*)(xp);
          const float4 x1 = *(const float4*)(xp + 4);
          a0 += wv * x0.x; a1 += wv * x0.y; a2 += wv * x0.z; a3 += wv * x0.w;
          a4 += wv * x1.x; a5 += wv * x1.y; a6 += wv * x1.z; a7 += wv * x1.w;
        }
        float* ap = &s_a[r * AST + kq];
        *(float4*)(ap)     = make_float4(a0, a1, a2, a3);
        *(float4*)(ap + 4) = make_float4(a4, a5, a6, a7);
      }
      __syncthreads();

      // ---- WMMA over this K chunk: 4 k-steps x (2 M-tiles x 4 N-tiles) ----
      #pragma unroll
      for (int kk = 0; kk < KCH; kk += 4) {
        v2f a0 = *(const v2f*)(&s_a[(rsg * 32 +      lm) * AST + kk + kh]);
        v2f a1 = *(const v2f*)(&s_a[(rsg * 32 + 16 + lm) * AST + kk + kh]);
        #pragma unroll
        for (int nt = 0; nt < 4; ++nt) {
          int col = ch * 64 + nt * 16 + lm;
          v2f bf;
          bf.x = s_b[(kk + kh)     * BST + col];
          bf.y = s_b[(kk + kh + 1) * BST + col];
          acc[0][nt] = __builtin_amdgcn_wmma_f32_16x16x4_f32(
              false, a0, false, bf, (short)0, acc[0][nt], false, false);
          acc[1][nt] = __builtin_amdgcn_wmma_f32_16x16x4_f32(
              false, a1, false, bf, (short)0, acc[1][nt], false, false);
        }
      }
    }
  }

  // ---- epilogue: bias + ELU + store ----
  const int trow_add = (lane >> 4) * 8;   // C/D layout: VGPR j -> row j (lanes 0-15) / j+8 (16-31)
  #pragma unroll
  for (int mt = 0; mt < 2; ++mt) {
    #pragma unroll
    for (int nt = 0; nt < 4; ++nt) {
      int col = ch * 64 + nt * 16 + lm;
      #pragma unroll
      for (int j = 0; j < 8; ++j) {
        int pl = mt * 16 + j + trow_add;
        int p  = pt0 + pl;
        float v = acc[mt][nt][j] + bias[p * CO + col];
        v = (v > 0.0f) ? v : expm1f(v);
        out[((rsg * PP + p) * CO) + col] = v;
      }
    }
  }
}

extern "C" void kernel_launch(void* const* d_in, const int* in_sizes, int n_in,
                              void* d_out, int out_size, void* d_ws, size_t ws_size,
                              hipStream_t stream) {
  const float* in_pc   = (const float*)d_in[0];
  const float* raw_w   = (const float*)d_in[1];
  const float* weights = (const float*)d_in[2];
  const float* bias    = (const float*)d_in[3];
  const int*   nbr     = (const int*)d_in[4];
  float*       out     = (float*)d_out;

  dim3 grid(PP / PTS);   // 625 blocks, each: 32 points x 4 batches x 128 outputs
  lasm_conv_wmma_kernel<<<grid, 256, 0, stream>>>(in_pc, raw_w, weights, bias, nbr, out);
}